// SparseLinear_89421219103267
// MI455X (gfx1250) — compile-verified
//
#include <hip/hip_runtime.h>

typedef __attribute__((ext_vector_type(2))) float v2f;
typedef __attribute__((ext_vector_type(8))) float v8f;

#define K_DIM   4096
#define N_DIM   4096
#define BM      128
#define BN      128
#define KB      32
#define SK      36          // padded LDS row stride (dwords): conflict-free, 16B-aligned
#define THRESH  1e-6f

// One wave-wide async copy: each lane moves 16B global -> LDS (ASYNCcnt-tracked).
__device__ __forceinline__ void async_copy_b128(unsigned lds_byte, const float* gaddr) {
    asm volatile("global_load_async_to_lds_b128 %0, %1, off"
                 :: "v"(lds_byte), "v"(gaddr) : "memory");
}
__device__ __forceinline__ void wait_async_zero() {
    asm volatile("s_wait_asynccnt 0x0" ::: "memory");
}

// y = (tile-masked x) @ W^T + b, fused tile-activity detection.
// Block: 256 threads = 8 wave32 waves (2 M x 4 N). Each wave: 4x2 tiles of 16x16.
// Double-buffered LDS fed by global_load_async_to_lds_b128.
__global__ __launch_bounds__(256)
void sparse_linear_wmma(const float* __restrict__ X,
                        const float* __restrict__ W,
                        const float* __restrict__ Bv,
                        float* __restrict__ Y)
{
    __shared__ float    lA[2][BM * SK];
    __shared__ float    lB[2][BN * SK];
    __shared__ unsigned sflag[2];

    const int tid = threadIdx.x;
    const int n0  = blockIdx.x * BN;
    const int m0  = blockIdx.y * BM;

    if (tid < 2) sflag[tid] = 0u;

    const int lane   = tid & 31;      // wave32
    const int wid    = tid >> 5;
    const int lhalf  = lane >> 4;     // 0 or 1
    const int l16    = lane & 15;

    const int waveM0 = (wid >> 2) * 64;   // 0 or 64
    const int waveN0 = (wid & 3) * 32;    // 0..96

    // Per-thread async-copy mapping: 4 x b128 per tile per thread.
    const int crow = tid >> 3;            // 0..31 (advances by 32 per i)
    const int ckv  = (tid & 7) * 4;       // 0..28

    v8f zero = {};
    v8f acc[4][2];
    #pragma unroll
    for (int tm = 0; tm < 4; ++tm)
        #pragma unroll
        for (int tn = 0; tn < 2; ++tn)
            acc[tm][tn] = zero;

    // Running max|x| over this wave's rows [waveM0, waveM0+64) x all K.
    // Single v_max_num_f32 (with |src| modifier) per element; one compare at end.
    float amax = 0.0f;

    // ---- prologue: async-load tile 0 into buffer 0 ----
    #pragma unroll
    for (int i = 0; i < 4; ++i) {
        const int row = crow + 32 * i;
        async_copy_b128((unsigned)(size_t)(&lA[0][row * SK + ckv]),
                        X + (size_t)(m0 + row) * K_DIM + ckv);
        async_copy_b128((unsigned)(size_t)(&lB[0][row * SK + ckv]),
                        W + (size_t)(n0 + row) * K_DIM + ckv);
    }
    wait_async_zero();
    __syncthreads();    // tile 0 resident everywhere; sflag initialized

    for (int kt = 0; kt < K_DIM; kt += KB) {
        const int cur = (kt / KB) & 1;
        const int nxt = cur ^ 1;

        // issue async loads for the next K-tile into the other buffer
        if (kt + KB < K_DIM) {
            #pragma unroll
            for (int i = 0; i < 4; ++i) {
                const int row = crow + 32 * i;
                async_copy_b128((unsigned)(size_t)(&lA[nxt][row * SK + ckv]),
                                X + (size_t)(m0 + row) * K_DIM + (kt + KB) + ckv);
                async_copy_b128((unsigned)(size_t)(&lB[nxt][row * SK + ckv]),
                                W + (size_t)(n0 + row) * K_DIM + (kt + KB) + ckv);
            }
        }

        // 16x16x4 f32 WMMA over the current K-tile.
        // A frag: lane l16 = row M, K pair (kk+2*lhalf, +1) -> one b64 LDS load
        // B frag: lane l16 = col N, same K pair             -> one b64 LDS load
        const float* __restrict__ bufA = lA[cur];
        const float* __restrict__ bufB = lB[cur];
        #pragma unroll
        for (int kk = 0; kk < KB; kk += 4) {
            const int kof = kk + 2 * lhalf;
            v2f afrag[4], bfrag[2];
            #pragma unroll
            for (int tm = 0; tm < 4; ++tm)
                afrag[tm] = *(const v2f*)(bufA + (waveM0 + 16 * tm + l16) * SK + kof);
            #pragma unroll
            for (int tn = 0; tn < 2; ++tn)
                bfrag[tn] = *(const v2f*)(bufB + (waveN0 + 16 * tn + l16) * SK + kof);

            // activity: union over lanes/tm/kk/kt covers this wave's full 64xK block
            #pragma unroll
            for (int tm = 0; tm < 4; ++tm) {
                amax = fmaxf(amax, fabsf(afrag[tm].x));
                amax = fmaxf(amax, fabsf(afrag[tm].y));
            }

            #pragma unroll
            for (int tm = 0; tm < 4; ++tm)
                #pragma unroll
                for (int tn = 0; tn < 2; ++tn)
                    acc[tm][tn] = __builtin_amdgcn_wmma_f32_16x16x4_f32(
                        /*neg_a=*/false, afrag[tm],
                        /*neg_b=*/false, bfrag[tn],
                        /*c_mod=*/(short)0, acc[tm][tn],
                        /*reuse_a=*/false, /*reuse_b=*/false);
        }

        // next tile landed (own asyncs) + all waves done reading cur
        wait_async_zero();
        __syncthreads();
    }

    // Reduce 64-row tile activity across the block (max>thr <=> any(|x|>thr)).
    if (amax > THRESH) atomicOr(&sflag[waveM0 >> 6], 1u);
    __syncthreads();
    const unsigned act0 = sflag[0];
    const unsigned act1 = sflag[1];

    // Epilogue: y = active ? acc + b : b   (exact reference semantics)
    #pragma unroll
    for (int tn = 0; tn < 2; ++tn) {
        const int   col  = n0 + waveN0 + 16 * tn + l16;
        const float bias = Bv[col];
        #pragma unroll
        for (int tm = 0; tm < 4; ++tm) {
            #pragma unroll
            for (int r = 0; r < 8; ++r) {
                const int row_local = waveM0 + 16 * tm + r + 8 * lhalf;
                const unsigned act  = (row_local < 64) ? act0 : act1;
                const float v = act ? (acc[tm][tn][r] + bias) : bias;
                Y[(size_t)(m0 + row_local) * N_DIM + col] = v;
            }
        }
    }
}

extern "C" void kernel_launch(void* const* d_in, const int* in_sizes, int n_in,
                              void* d_out, int out_size, void* d_ws, size_t ws_size,
                              hipStream_t stream) {
    const float* X  = (const float*)d_in[0];
    const float* W  = (const float*)d_in[1];
    const float* Bv = (const float*)d_in[2];
    float* Y = (float*)d_out;

    const int Mrows = in_sizes[0] / K_DIM;        // 16384
    dim3 grid(N_DIM / BN, Mrows / BM);            // (32, 128)
    sparse_linear_wmma<<<grid, dim3(256, 1, 1), 0, stream>>>(X, W, Bv, Y);
}